// PCANet_88914412961936
// MI455X (gfx1250) — compile-verified
//
#include <hip/hip_runtime.h>
#include <math.h>

// ---------------- problem constants ----------------
#define NN    20000          // nodes
#define FF    64             // features
#define DEG   16             // in-degree incl self loop
#define CC    16             // classes
#define DF    (DEG * FF)     // 1024 = neigh row width
#define PITER 16             // subspace-iteration steps

typedef float v2f __attribute__((ext_vector_type(2)));
typedef float v8f __attribute__((ext_vector_type(8)));
typedef int   v4i_b __attribute__((vector_size(16)));   // matches builtin param type

#define AS1 __attribute__((address_space(1)))
#define AS3 __attribute__((address_space(3)))

#if __has_builtin(__builtin_amdgcn_global_load_async_to_lds_b128)
#define HAVE_ASYNC_LDS 1
#else
#define HAVE_ASYNC_LDS 0
#endif

// =====================================================================
// Generic fp32 WMMA GEMM: C[M,N] = A[M,K] @ op(B) (+bias) (+relu)
// A row-major (lda), B row-major (ldb); TRANSB: op(B)[k][n] = B[n][k].
// One 16x16 output tile per wave, 8 waves / block. M,N mult of 16; K mult of 4.
// TRANSB is a template parameter so the K-loop has no runtime branch.
// =====================================================================
template <int TRANSB>
__global__ __launch_bounds__(256)
void gemm_wmma_f32(const float* __restrict__ A, int lda,
                   const float* __restrict__ B, int ldb,
                   float* __restrict__ C, int ldc,
                   int M, int N, int K,
                   const float* __restrict__ bias, int relu)
{
    const int lane = threadIdx.x & 31;
    const int wave = threadIdx.x >> 5;
    const int tilesN = N >> 4;
    const int totalTiles = (M >> 4) * tilesN;
    const int tile = blockIdx.x * 8 + wave;
    if (tile >= totalTiles) return;               // wave-uniform: EXEC stays full

    const int ti = tile / tilesN;
    const int tj = tile - ti * tilesN;
    const int mbase = ti << 4;
    const int nbase = tj << 4;
    const int m16   = lane & 15;                  // A row within tile / B col within tile
    const int khalf = (lane >> 4) << 1;           // 0 or 2 (ISA 16x4 f32 A layout)

    const float* arow = A + (size_t)(mbase + m16) * lda + khalf;
    const float* brow = TRANSB ? (B + (size_t)(nbase + m16) * ldb + khalf)
                               : (B + nbase + m16);

    v8f acc = {0.f,0.f,0.f,0.f,0.f,0.f,0.f,0.f};

    #pragma unroll 4
    for (int k0 = 0; k0 < K; k0 += 4) {
        v2f a, b;
        a.x = arow[k0];
        a.y = arow[k0 + 1];
        if (TRANSB) {
            b.x = brow[k0];
            b.y = brow[k0 + 1];
        } else {
            b.x = brow[(size_t)(k0 + khalf    ) * ldb];
            b.y = brow[(size_t)(k0 + khalf + 1) * ldb];
        }
        acc = __builtin_amdgcn_wmma_f32_16x16x4_f32(false, a, false, b,
                                                    (short)0, acc, false, false);
    }

    const int rowoff = (lane >> 4) << 3;          // D layout: lanes>=16 hold M=8..15
    #pragma unroll
    for (int r = 0; r < 8; ++r) {
        const int row = mbase + rowoff + r;
        const int col = nbase + m16;
        float v = acc[r];
        if (bias) v += bias[col];
        if (relu) v = fmaxf(v, 0.f);
        C[(size_t)row * ldc + col] = v;
    }
}

// =====================================================================
// Gram with fused rank-1 centering: G = Xᵀ X - N·μμᵀ,  X = neigh [NN x 1024]
// Block computes 64x64 tile of G; 32-row K-slabs of both operands staged in
// LDS via GLOBAL_LOAD_ASYNC_TO_LDS_B128 (ASYNCcnt) when available.
// 8 waves x 2 subtiles each = 16 16x16 subtiles.
// =====================================================================
__global__ __launch_bounds__(256)
void gram_wmma(const float* __restrict__ X,
               const float* __restrict__ mu,
               float* __restrict__ G, int Nrows)
{
    __shared__ float As[32][68];
    __shared__ float Bs[32][68];

    const int ibase = blockIdx.y * 64;
    const int jbase = blockIdx.x * 64;
    const int lane  = threadIdx.x & 31;
    const int wave  = threadIdx.x >> 5;
    const int m16   = lane & 15;
    const int khalf = (lane >> 4) << 1;

    const int s0 = wave, s1 = wave + 8;           // two subtiles per wave
    const int ti0 = s0 >> 2, tj0 = s0 & 3;
    const int ti1 = s1 >> 2, tj1 = s1 & 3;

    v8f acc0 = {0.f,0.f,0.f,0.f,0.f,0.f,0.f,0.f};
    v8f acc1 = {0.f,0.f,0.f,0.f,0.f,0.f,0.f,0.f};

    for (int k0 = 0; k0 < Nrows; k0 += 32) {
        #pragma unroll
        for (int i = 0; i < 2; ++i) {
            const int chunk = threadIdx.x + i * 256;     // 0..511 float4-chunks
            const int r  = chunk >> 4;                   // 0..31
            const int c4 = (chunk & 15) << 2;            // 0..60, step 4
            const float* gA = X + (size_t)(k0 + r) * DF + ibase + c4;
            const float* gB = X + (size_t)(k0 + r) * DF + jbase + c4;
#if HAVE_ASYNC_LDS
            __builtin_amdgcn_global_load_async_to_lds_b128(
                (AS1 v4i_b*)(unsigned long long)gA, (AS3 v4i_b*)&As[r][c4], 0, 0);
            __builtin_amdgcn_global_load_async_to_lds_b128(
                (AS1 v4i_b*)(unsigned long long)gB, (AS3 v4i_b*)&Bs[r][c4], 0, 0);
#else
            *(float4*)&As[r][c4] = *(const float4*)gA;
            *(float4*)&Bs[r][c4] = *(const float4*)gB;
#endif
        }
#if HAVE_ASYNC_LDS
        asm volatile("s_wait_asynccnt 0" ::: "memory");
#endif
        __syncthreads();
        #pragma unroll
        for (int kk = 0; kk < 32; kk += 4) {
            v2f a0, b0, a1, b1;
            a0.x = As[kk + khalf    ][ti0*16 + m16];
            a0.y = As[kk + khalf + 1][ti0*16 + m16];
            b0.x = Bs[kk + khalf    ][tj0*16 + m16];
            b0.y = Bs[kk + khalf + 1][tj0*16 + m16];
            acc0 = __builtin_amdgcn_wmma_f32_16x16x4_f32(false, a0, false, b0,
                                                         (short)0, acc0, false, false);
            a1.x = As[kk + khalf    ][ti1*16 + m16];
            a1.y = As[kk + khalf + 1][ti1*16 + m16];
            b1.x = Bs[kk + khalf    ][tj1*16 + m16];
            b1.y = Bs[kk + khalf + 1][tj1*16 + m16];
            acc1 = __builtin_amdgcn_wmma_f32_16x16x4_f32(false, a1, false, b1,
                                                         (short)0, acc1, false, false);
        }
        __syncthreads();
    }

    const int rowoff = (lane >> 4) << 3;
    const float nN = (float)Nrows;
    #pragma unroll
    for (int r = 0; r < 8; ++r) {
        int grow = ibase + ti0*16 + rowoff + r;
        int gcol = jbase + tj0*16 + m16;
        G[(size_t)grow * DF + gcol] = acc0[r] - nN * mu[grow] * mu[gcol];
        grow = ibase + ti1*16 + rowoff + r;
        gcol = jbase + tj1*16 + m16;
        G[(size_t)grow * DF + gcol] = acc1[r] - nN * mu[grow] * mu[gcol];
    }
}

// =====================================================================
// neigh[n, k*64+f] = (1/16) * hl[src(n,k), f]; src(n,15)=n (self loop last,
// matching PyG stable-sort-by-dst packing). 16 slots per block, float4 copies.
// =====================================================================
__global__ __launch_bounds__(256)
void build_neigh(const float* __restrict__ hl,
                 const int* __restrict__ src,
                 float* __restrict__ neigh)
{
    const int t    = threadIdx.x;
    const int slot = blockIdx.x * 16 + (t >> 4);    // over NN*16 node-slot pairs
    const int n = slot >> 4;
    const int k = slot & 15;
    const int srow = (k < 15) ? src[n * 15 + k] : n;
    const int f4 = (t & 15) << 2;
    const float4 v = *(const float4*)(hl + (size_t)srow * FF + f4);
    float4 o;
    o.x = v.x * 0.0625f; o.y = v.y * 0.0625f;
    o.z = v.z * 0.0625f; o.w = v.w * 0.0625f;
    *(float4*)(neigh + (size_t)n * DF + (k << 6) + f4) = o;
}

// Column means of neigh: block b owns 64 columns, 4 row-groups of 64 threads.
__global__ __launch_bounds__(256)
void colmean(const float* __restrict__ X, float* __restrict__ mu)
{
    __shared__ float red[256];
    const int t   = threadIdx.x;
    const int col = blockIdx.x * 64 + (t & 63);
    float s = 0.f;
    for (int n = (t >> 6); n < NN; n += 4)
        s += X[(size_t)n * DF + col];
    red[t] = s;
    __syncthreads();
    if (t < 64)
        mu[blockIdx.x * 64 + t] =
            (red[t] + red[t + 64] + red[t + 128] + red[t + 192]) * (1.f / (float)NN);
}

// Deterministic pseudo-random init for subspace basis V [1024 x 64]
__global__ void init_v(float* __restrict__ V)
{
    const int idx = blockIdx.x * 256 + threadIdx.x;
    if (idx < DF * FF) {
        unsigned h = (unsigned)idx * 2654435761u;
        h ^= h >> 16; h *= 0x85ebca6bu; h ^= h >> 13; h *= 0xc2b2ae35u; h ^= h >> 16;
        V[idx] = ((float)(h & 0xFFFFFFu) * (1.f / 8388608.f)) - 1.f;
    }
}

// =====================================================================
// Modified Gram-Schmidt: Y [1024 x 64] -> orthonormal columns written to V.
// One block of 1024 threads (32 waves); column j normalized, waves strip-mine
// the trailing-column updates with wave32 shuffle reductions.
// =====================================================================
__global__ __launch_bounds__(1024)
void mgs_kernel(float* __restrict__ Y, float* __restrict__ V)
{
    __shared__ float cj[1024];
    __shared__ float red[1024];
    const int t = threadIdx.x;
    const int lane = t & 31, wave = t >> 5;

    for (int j = 0; j < 64; ++j) {
        float v = Y[(size_t)t * 64 + j];
        red[t] = v * v;
        __syncthreads();
        for (int s = 512; s > 0; s >>= 1) {
            if (t < s) red[t] += red[t + s];
            __syncthreads();
        }
        const float rinv = rsqrtf(red[0] + 1e-20f);
        const float vn = v * rinv;
        cj[t] = vn;
        V[(size_t)t * 64 + j] = vn;
        __syncthreads();
        for (int k = j + 1 + wave; k < 64; k += 32) {
            float p = 0.f;
            for (int r = lane; r < 1024; r += 32)
                p += cj[r] * Y[(size_t)r * 64 + k];
            for (int off = 16; off > 0; off >>= 1)
                p += __shfl_down(p, off, 32);
            const float dot = __shfl(p, 0, 32);
            for (int r = lane; r < 1024; r += 32)
                Y[(size_t)r * 64 + k] -= dot * cj[r];
        }
        __syncthreads();
    }
}

// T = Vᵀ (G V)  -> 64x64 (small, scalar)
__global__ __launch_bounds__(256)
void vtgv(const float* __restrict__ V, const float* __restrict__ GV,
          float* __restrict__ T)
{
    const int idx = blockIdx.x * 256 + threadIdx.x;
    if (idx >= 64 * 64) return;
    const int i = idx >> 6, j = idx & 63;
    float s = 0.f;
    for (int r = 0; r < 1024; ++r)
        s += V[r * 64 + i] * GV[r * 64 + j];
    T[idx] = s;
}

// =====================================================================
// Cyclic Jacobi eigensolve of 64x64 symmetric T; outputs eigenvector matrix
// with columns sorted by descending eigenvalue (Rayleigh-Ritz rotation).
// One block of 64 threads.
// =====================================================================
__global__ __launch_bounds__(64)
void jacobi64(const float* __restrict__ Tin, float* __restrict__ Qout)
{
    __shared__ float A[64][65];
    __shared__ float Q[64][65];
    __shared__ float cs[2];
    __shared__ int order[64];
    const int t = threadIdx.x;

    for (int r = 0; r < 64; ++r) A[r][t] = Tin[r * 64 + t];
    for (int r = 0; r < 64; ++r) Q[r][t] = (r == t) ? 1.f : 0.f;
    __syncthreads();

    for (int sweep = 0; sweep < 8; ++sweep) {
        for (int p = 0; p < 63; ++p) {
            for (int q = p + 1; q < 64; ++q) {
                if (t == 0) {
                    const float apq = A[p][q];
                    float c = 1.f, s = 0.f;
                    if (fabsf(apq) > 1e-12f) {
                        const float tau = (A[q][q] - A[p][p]) / (2.f * apq);
                        const float tt = (tau >= 0.f ? 1.f : -1.f) /
                                         (fabsf(tau) + sqrtf(1.f + tau * tau));
                        c = rsqrtf(1.f + tt * tt);
                        s = tt * c;
                    }
                    cs[0] = c; cs[1] = s;
                }
                __syncthreads();
                const float c = cs[0], s = cs[1];
                // rotate columns p,q of A and Q (thread t = row)
                float ap = A[t][p], aq = A[t][q];
                A[t][p] = c * ap - s * aq;
                A[t][q] = s * ap + c * aq;
                const float qp = Q[t][p], qq = Q[t][q];
                Q[t][p] = c * qp - s * qq;
                Q[t][q] = s * qp + c * qq;
                __syncthreads();
                // rotate rows p,q of A (thread t = column)
                ap = A[p][t]; aq = A[q][t];
                A[p][t] = c * ap - s * aq;
                A[q][t] = s * ap + c * aq;
                __syncthreads();
            }
        }
    }

    if (t == 0) {
        bool used[64];
        for (int i = 0; i < 64; ++i) used[i] = false;
        for (int c = 0; c < 64; ++c) {
            int best = 0; float bv = -3.0e38f;
            for (int i = 0; i < 64; ++i)
                if (!used[i] && A[i][i] > bv) { bv = A[i][i]; best = i; }
            used[best] = true;
            order[c] = best;
        }
    }
    __syncthreads();
    for (int r = 0; r < 64; ++r) Qout[r * 64 + t] = Q[r][order[t]];
}

// Row-wise log_softmax over 16 classes (in place), 16 rows / block.
__global__ __launch_bounds__(256)
void logsoftmax16(float* __restrict__ out, int Nrows)
{
    const int t = threadIdx.x;
    const int row = blockIdx.x * 16 + (t >> 4);
    const int c = t & 15;
    if (row >= Nrows) return;
    const float x = out[(size_t)row * CC + c];
    float m = x;
    for (int off = 8; off > 0; off >>= 1) m = fmaxf(m, __shfl_xor(m, off, 16));
    float e = expf(x - m);
    for (int off = 8; off > 0; off >>= 1) e += __shfl_xor(e, off, 16);
    out[(size_t)row * CC + c] = x - m - logf(e);
}

// =====================================================================
// host-side orchestration
// =====================================================================
static inline void gemm(const float* A, int lda, const float* B, int ldb, int transB,
                        float* C, int ldc, int M, int N, int K,
                        const float* bias, int relu, hipStream_t stream)
{
    const int tiles = (M >> 4) * (N >> 4);
    const int blocks = (tiles + 7) / 8;
    if (transB)
        gemm_wmma_f32<1><<<blocks, 256, 0, stream>>>(A, lda, B, ldb, C, ldc,
                                                     M, N, K, bias, relu);
    else
        gemm_wmma_f32<0><<<blocks, 256, 0, stream>>>(A, lda, B, ldb, C, ldc,
                                                     M, N, K, bias, relu);
}

extern "C" void kernel_launch(void* const* d_in, const int* in_sizes, int n_in,
                              void* d_out, int out_size, void* d_ws, size_t ws_size,
                              hipStream_t stream)
{
    (void)in_sizes; (void)n_in; (void)out_size; (void)ws_size;

    const float* x   = (const float*)d_in[0];
    const int*   src = (const int*)  d_in[1];        // edge_index[0]; dst implicit
    const float* Ws[3]  = {(const float*)d_in[2], (const float*)d_in[4], (const float*)d_in[6]};
    const float* bs[3]  = {(const float*)d_in[3], (const float*)d_in[5], (const float*)d_in[7]};
    const float* Wfc = (const float*)d_in[8];
    const float* bfc = (const float*)d_in[9];
    float* out = (float*)d_out;

    // workspace carve (fp32)
    float* p = (float*)d_ws;
    float* h     = p; p += (size_t)NN * FF;          // activations
    float* work  = p; p += (size_t)NN * FF;          // hl / proj
    float* neigh = p; p += (size_t)NN * DF;          // 82 MB, L2-resident
    float* mu    = p; p += DF;
    float* G     = p; p += (size_t)DF * DF;          // 4 MB
    float* V     = p; p += (size_t)DF * FF;
    float* Y     = p; p += (size_t)DF * FF;          // also GV
    float* T64   = p; p += FF * FF;
    float* Q     = p; p += FF * FF;
    float* Vt    = p; p += (size_t)DF * FF;

    const float* lin = x;
    for (int layer = 0; layer < 3; ++layer) {
        // 1) hl = h @ Wᵀ
        gemm(lin, FF, Ws[layer], FF, /*transB=*/1, work, FF,
             NN, FF, FF, nullptr, 0, stream);
        // 2) gather messages (norm == 1/16 exactly: regular graph)
        build_neigh<<<NN, 256, 0, stream>>>(work, src, neigh);
        // 3) column means (for Gram centering)
        colmean<<<16, 256, 0, stream>>>(neigh, mu);
        // 4) centered Gram G = neighᵀ·neigh - N μμᵀ   (dominant WMMA GEMM)
        gram_wmma<<<dim3(16, 16), 256, 0, stream>>>(neigh, mu, G, NN);
        // 5) top-64 eigenspace: subspace iteration + MGS
        init_v<<<(DF * FF + 255) / 256, 256, 0, stream>>>(V);
        for (int it = 0; it < PITER; ++it) {
            gemm(G, DF, V, FF, 0, Y, FF, DF, FF, DF, nullptr, 0, stream);
            mgs_kernel<<<1, 1024, 0, stream>>>(Y, V);
        }
        // 6) Rayleigh-Ritz: T = Vᵀ G V; eigendecompose; Vtop = V Q (desc order)
        gemm(G, DF, V, FF, 0, Y, FF, DF, FF, DF, nullptr, 0, stream);   // Y = G V
        vtgv<<<16, 256, 0, stream>>>(V, Y, T64);
        jacobi64<<<1, 64, 0, stream>>>(T64, Q);
        gemm(V, FF, Q, FF, 0, Vt, FF, DF, FF, FF, nullptr, 0, stream);
        // 7) proj = neigh @ Vtop  (projects UNcentered neigh, as reference)
        gemm(neigh, DF, Vt, FF, 0, work, FF, NN, FF, DF, nullptr, 0, stream);
        // 8) out = relu(proj @ Wᵀ + b)
        gemm(work, FF, Ws[layer], FF, 1, h, FF, NN, FF, FF, bs[layer], 1, stream);
        lin = h;
    }

    // FC head + log_softmax
    gemm(h, FF, Wfc, FF, 1, out, CC, NN, CC, FF, bfc, 0, stream);
    logsoftmax16<<<NN / 16, 256, 0, stream>>>(out, NN);
}